// LevitStage_23665269801105
// MI455X (gfx1250) — compile-verified
//
#include <hip/hip_runtime.h>
#include <cstddef>
#include <cstdint>

// ---------------------------------------------------------------------------
// LeViT stage for MI455X (gfx1250). fp32 end-to-end.
// Dense GEMMs run on V_WMMA_F32_16X16X4_F32 with double-buffered LDS tiles
// filled by CDNA5 async copies (global_load_async_to_lds_b128 / ASYNCcnt).
// Attention core (softmax + relative-bias gather) is VALU + LDS broadcast.
// BN folded into each linear on device.
// ---------------------------------------------------------------------------

typedef __attribute__((ext_vector_type(2))) float v2f;
typedef __attribute__((ext_vector_type(8))) float v8f;

static constexpr int BATCH   = 128;
static constexpr int RES     = 14;
static constexpr int NTOK    = 196;
static constexpr int HID     = 256;
static constexpr int HEADS   = 4;
static constexpr int VDIM    = 64;       // RATIO*KD
static constexpr int QKV_OUT = 512;
static constexpr int SHEADS  = 8;
static constexpr int SKD     = 32;
static constexpr int SVD     = 64;
static constexpr int KV_OUT  = 768;
static constexpr int RES_OUT = 7;
static constexpr int NOUT    = 49;
static constexpr int HID2    = 384;
static constexpr int MT      = BATCH * NTOK;   // 25088
static constexpr int MT2     = BATCH * NOUT;   // 6272
static constexpr float SCALE = 0.17677669529663687f; // 32^-0.5

__device__ __forceinline__ int iabs_(int x) { return x < 0 ? -x : x; }
__device__ __forceinline__ float hswish_(float x) {
  float t = fminf(fmaxf(x + 3.0f, 0.0f), 6.0f);
  return x * t * (1.0f / 6.0f);
}

// CDNA5 async copy: 16B global -> LDS per lane, tracked by ASYNCcnt.
// Low 32 bits of a generic LDS pointer are the LDS byte offset (ISA 10.2).
__device__ __forceinline__ void async_b128(const float* g, float* l) {
  unsigned lds = (unsigned)(uintptr_t)l;
  asm volatile("global_load_async_to_lds_b128 %0, %1, off"
               :: "v"(lds), "v"(g) : "memory");
}
__device__ __forceinline__ void wait_async0() {
  asm volatile("s_wait_asynccnt 0" ::: "memory");
}

// ---------------------------------------------------------------------------
// Fold BN into linear.
// ---------------------------------------------------------------------------
__global__ void fuse_bn(const float* __restrict__ W, const float* __restrict__ g,
                        const float* __restrict__ b, const float* __restrict__ m,
                        const float* __restrict__ v, float* __restrict__ Wf,
                        float* __restrict__ beta, int dout, int din) {
  int idx = blockIdx.x * 256 + threadIdx.x;
  if (idx >= dout * din) return;
  int o = idx / din;
  float inv = g[o] / sqrtf(v[o] + 1e-5f);
  Wf[idx] = W[idx] * inv;
  if (idx - o * din == 0) beta[o] = b[o] - m[o] * inv;
}

// ---------------------------------------------------------------------------
// D[M,N] = act(A)[M,K] @ Wf[N,K]^T + beta[N] (+R).  M%64==0, N%128==0, K%32==0
// Block 256 thr = 8 waves; block tile 64x128; wave tile 32x32 (2x2 WMMA 16x16).
// Double-buffered LDS, async b128 fills. Hardswish applied at LDS-read time.
// ---------------------------------------------------------------------------
static constexpr int AST = 36;                // padded k-stride (16B aligned rows)
static constexpr int ABUF = 64 * AST;         // A tile floats
static constexpr int BBUF = 128 * AST;        // B tile floats
static constexpr int BUFSZ = ABUF + BBUF;

template <bool HSW, bool RESID>
__global__ __launch_bounds__(256) void gemm_bn(
    const float* __restrict__ A, const float* __restrict__ Wf,
    const float* __restrict__ beta, const float* __restrict__ R,
    float* __restrict__ D, int M, int N, int K) {
  __shared__ float sh[2 * BUFSZ];
  const int tid  = threadIdx.x;
  const int lane = tid & 31;
  const int w    = tid >> 5;
  const int wm   = (w & 1) * 32;        // 2 waves over M (64)
  const int wn   = (w >> 1) * 32;       // 4 waves over N (128)
  const int rowBase = blockIdx.x * 64;
  const int colBase = blockIdx.y * 128;

  v8f a00 = {0.f,0.f,0.f,0.f,0.f,0.f,0.f,0.f};
  v8f a01 = a00, a10 = a00, a11 = a00;

  const int nc = K >> 5;
  // --- prologue: fill buffer 0 -------------------------------------------
  {
    float* As = sh;  float* Bs = sh + ABUF;
#pragma unroll
    for (int i = 0; i < 2; ++i) {       // A: 64x32 = 512 f4
      int e = i * 256 + tid, mm = e >> 3, q4 = e & 7;
      async_b128(A + (size_t)(rowBase + mm) * K + q4 * 4, &As[mm * AST + q4 * 4]);
    }
#pragma unroll
    for (int i = 0; i < 4; ++i) {       // B: 128x32 = 1024 f4
      int e = i * 256 + tid, nn = e >> 3, q4 = e & 7;
      async_b128(Wf + (size_t)(colBase + nn) * K + q4 * 4, &Bs[nn * AST + q4 * 4]);
    }
  }
  wait_async0();
  __syncthreads();

  int p = 0;
  for (int c = 0; c < nc; ++c) {
    if (c + 1 < nc) {                   // overlap next-tile fill with compute
      const int k0 = (c + 1) * 32;
      float* As = sh + (p ^ 1) * BUFSZ; float* Bs = As + ABUF;
#pragma unroll
      for (int i = 0; i < 2; ++i) {
        int e = i * 256 + tid, mm = e >> 3, q4 = e & 7;
        async_b128(A + (size_t)(rowBase + mm) * K + k0 + q4 * 4,
                   &As[mm * AST + q4 * 4]);
      }
#pragma unroll
      for (int i = 0; i < 4; ++i) {
        int e = i * 256 + tid, nn = e >> 3, q4 = e & 7;
        async_b128(Wf + (size_t)(colBase + nn) * K + k0 + q4 * 4,
                   &Bs[nn * AST + q4 * 4]);
      }
    }
    // --- compute on buffer p ---------------------------------------------
    {
      const float* As = sh + p * BUFSZ;
      const float* Bs = As + ABUF;
      const int l15 = lane & 15;
      const int kb  = (lane >> 4) * 2;
#pragma unroll
      for (int kk = 0; kk < 32; kk += 4) {
        v2f fa0 = *(const v2f*)&As[(wm + l15) * AST + kk + kb];
        v2f fa1 = *(const v2f*)&As[(wm + 16 + l15) * AST + kk + kb];
        v2f fb0 = *(const v2f*)&Bs[(wn + l15) * AST + kk + kb];
        v2f fb1 = *(const v2f*)&Bs[(wn + 16 + l15) * AST + kk + kb];
        if (HSW) {
          fa0.x = hswish_(fa0.x); fa0.y = hswish_(fa0.y);
          fa1.x = hswish_(fa1.x); fa1.y = hswish_(fa1.y);
        }
        a00 = __builtin_amdgcn_wmma_f32_16x16x4_f32(false, fa0, false, fb0,
                                                    (short)0, a00, false, false);
        a01 = __builtin_amdgcn_wmma_f32_16x16x4_f32(false, fa0, false, fb1,
                                                    (short)0, a01, false, false);
        a10 = __builtin_amdgcn_wmma_f32_16x16x4_f32(false, fa1, false, fb0,
                                                    (short)0, a10, false, false);
        a11 = __builtin_amdgcn_wmma_f32_16x16x4_f32(false, fa1, false, fb1,
                                                    (short)0, a11, false, false);
      }
    }
    wait_async0();
    __syncthreads();
    p ^= 1;
  }

  // --- epilogue -----------------------------------------------------------
  const int half = lane >> 4;
  const int l15  = lane & 15;
  const int nc0  = colBase + wn + l15;
  const int nc1  = nc0 + 16;
  const float bt0 = beta[nc0], bt1 = beta[nc1];
#pragma unroll
  for (int r = 0; r < 8; ++r) {
    int mr0 = rowBase + wm + r + 8 * half;
    int mr1 = mr0 + 16;
    float d00 = a00[r] + bt0, d01 = a01[r] + bt1;
    float d10 = a10[r] + bt0, d11 = a11[r] + bt1;
    if (RESID) {
      d00 += R[(size_t)mr0 * N + nc0]; d01 += R[(size_t)mr0 * N + nc1];
      d10 += R[(size_t)mr1 * N + nc0]; d11 += R[(size_t)mr1 * N + nc1];
    }
    D[(size_t)mr0 * N + nc0] = d00; D[(size_t)mr0 * N + nc1] = d01;
    D[(size_t)mr1 * N + nc0] = d10; D[(size_t)mr1 * N + nc1] = d11;
  }
}

// ---------------------------------------------------------------------------
// Main attention: scores + softmax (unnormalized probs; rowsum saved).
// grid = B*HEADS, block 256 (196 active rows). bias id = |di|*14 + |dj|.
// ---------------------------------------------------------------------------
__global__ __launch_bounds__(256) void attn_scores_main(
    const float* __restrict__ qkv, const float* __restrict__ bias,
    float* __restrict__ S, float* __restrict__ rowsum) {
  const int bh = blockIdx.x, b = bh / HEADS, h = bh % HEADS;
  __shared__ float Ks[NTOK][36];
  for (int e = threadIdx.x; e < NTOK * 8; e += 256) {
    int mm = e >> 3, q4 = e & 7;
    *(float4*)&Ks[mm][q4 * 4] =
        *(const float4*)&qkv[(size_t)(b * NTOK + mm) * QKV_OUT + h * 128 + 32 + q4 * 4];
  }
  __syncthreads();
  const int n = threadIdx.x;
  if (n >= NTOK) return;
  float4 q4r[8];
#pragma unroll
  for (int i = 0; i < 8; ++i)
    q4r[i] = *(const float4*)&qkv[(size_t)(b * NTOK + n) * QKV_OUT + h * 128 + i * 4];
  const int i1 = n / RES, j1 = n - i1 * RES;
  const size_t srow = ((size_t)bh * NTOK + n) * NTOK;
  float smax = -1e30f;
  for (int m = 0; m < NTOK; ++m) {
    float s = 0.f;
#pragma unroll
    for (int i = 0; i < 8; ++i) {
      float4 kv = *(const float4*)&Ks[m][i * 4];
      s += q4r[i].x * kv.x + q4r[i].y * kv.y + q4r[i].z * kv.z + q4r[i].w * kv.w;
    }
    int i2 = m / RES, j2 = m - i2 * RES;
    s = s * SCALE + bias[h * 196 + iabs_(i1 - i2) * RES + iabs_(j1 - j2)];
    S[srow + m] = s;
    smax = fmaxf(smax, s);
  }
  float sum = 0.f;
  for (int m = 0; m < NTOK; ++m) {
    float pp = __expf(S[srow + m] - smax);
    S[srow + m] = pp;
    sum += pp;
  }
  rowsum[(size_t)bh * NTOK + n] = sum;
}

__global__ __launch_bounds__(256) void attn_av_main(
    const float* __restrict__ qkv, const float* __restrict__ S,
    const float* __restrict__ rowsum, float* __restrict__ O) {
  const int bh = blockIdx.x, b = bh / HEADS, h = bh % HEADS;
  __shared__ float Vs[NTOK][68];
  for (int e = threadIdx.x; e < NTOK * 16; e += 256) {
    int mm = e >> 4, q4 = e & 15;
    *(float4*)&Vs[mm][q4 * 4] =
        *(const float4*)&qkv[(size_t)(b * NTOK + mm) * QKV_OUT + h * 128 + 64 + q4 * 4];
  }
  __syncthreads();
  const int n = threadIdx.x;
  if (n >= NTOK) return;
  float4 acc[16];
#pragma unroll
  for (int i = 0; i < 16; ++i) acc[i] = make_float4(0.f, 0.f, 0.f, 0.f);
  const size_t srow = ((size_t)bh * NTOK + n) * NTOK;
  for (int m = 0; m < NTOK; ++m) {
    float pp = S[srow + m];
#pragma unroll
    for (int i = 0; i < 16; ++i) {
      float4 vv = *(const float4*)&Vs[m][i * 4];
      acc[i].x += pp * vv.x; acc[i].y += pp * vv.y;
      acc[i].z += pp * vv.z; acc[i].w += pp * vv.w;
    }
  }
  const float inv = 1.0f / rowsum[(size_t)bh * NTOK + n];
  float* orow = &O[(size_t)(b * NTOK + n) * (HEADS * VDIM) + h * VDIM];
#pragma unroll
  for (int i = 0; i < 16; ++i) {
    float4 o4 = make_float4(acc[i].x * inv, acc[i].y * inv,
                            acc[i].z * inv, acc[i].w * inv);
    *(float4*)&orow[i * 4] = o4;
  }
}

// Gather strided queries (float4 wide).
__global__ void gather_q(const float* __restrict__ X, float* __restrict__ XQ) {
  int idx = blockIdx.x * 256 + threadIdx.x;
  if (idx >= MT2 * (HID / 4)) return;
  int c4 = idx & 63;
  int t = idx >> 6;
  int b = t / NOUT, n = t - b * NOUT;
  int i = (n / RES_OUT) * 2, j = (n % RES_OUT) * 2;
  ((float4*)XQ)[idx] =
      ((const float4*)X)[((size_t)b * NTOK + i * RES + j) * 64 + c4];
}

// Sub-attention scores+softmax. grid = B*SHEADS, block 64 (49 active rows).
__global__ __launch_bounds__(64) void attn_scores_sub(
    const float* __restrict__ Q, const float* __restrict__ KV,
    const float* __restrict__ bias, float* __restrict__ S,
    float* __restrict__ rowsum) {
  const int bh = blockIdx.x, b = bh / SHEADS, h = bh % SHEADS;
  __shared__ float Ks[NTOK][36];
  for (int e = threadIdx.x; e < NTOK * 8; e += 64) {
    int mm = e >> 3, q4 = e & 7;
    *(float4*)&Ks[mm][q4 * 4] =
        *(const float4*)&KV[(size_t)(b * NTOK + mm) * KV_OUT + h * 96 + q4 * 4];
  }
  __syncthreads();
  const int n = threadIdx.x;
  if (n >= NOUT) return;
  float4 q4r[8];
#pragma unroll
  for (int i = 0; i < 8; ++i)
    q4r[i] = *(const float4*)&Q[(size_t)(b * NOUT + n) * (SHEADS * SKD) + h * SKD + i * 4];
  const int i1 = (n / RES_OUT) * 2, j1 = (n % RES_OUT) * 2;
  const size_t srow = ((size_t)bh * NOUT + n) * NTOK;
  float smax = -1e30f;
  for (int m = 0; m < NTOK; ++m) {
    float s = 0.f;
#pragma unroll
    for (int i = 0; i < 8; ++i) {
      float4 kv = *(const float4*)&Ks[m][i * 4];
      s += q4r[i].x * kv.x + q4r[i].y * kv.y + q4r[i].z * kv.z + q4r[i].w * kv.w;
    }
    int i2 = m / RES, j2 = m - i2 * RES;
    s = s * SCALE + bias[h * 196 + iabs_(i1 - i2) * RES + iabs_(j1 - j2)];
    S[srow + m] = s;
    smax = fmaxf(smax, s);
  }
  float sum = 0.f;
  for (int m = 0; m < NTOK; ++m) {
    float pp = __expf(S[srow + m] - smax);
    S[srow + m] = pp;
    sum += pp;
  }
  rowsum[(size_t)bh * NOUT + n] = sum;
}

__global__ __launch_bounds__(64) void attn_av_sub(
    const float* __restrict__ KV, const float* __restrict__ S,
    const float* __restrict__ rowsum, float* __restrict__ O) {
  const int bh = blockIdx.x, b = bh / SHEADS, h = bh % SHEADS;
  __shared__ float Vs[NTOK][68];
  for (int e = threadIdx.x; e < NTOK * 16; e += 64) {
    int mm = e >> 4, q4 = e & 15;
    *(float4*)&Vs[mm][q4 * 4] =
        *(const float4*)&KV[(size_t)(b * NTOK + mm) * KV_OUT + h * 96 + 32 + q4 * 4];
  }
  __syncthreads();
  const int n = threadIdx.x;
  if (n >= NOUT) return;
  float4 acc[16];
#pragma unroll
  for (int i = 0; i < 16; ++i) acc[i] = make_float4(0.f, 0.f, 0.f, 0.f);
  const size_t srow = ((size_t)bh * NOUT + n) * NTOK;
  for (int m = 0; m < NTOK; ++m) {
    float pp = S[srow + m];
#pragma unroll
    for (int i = 0; i < 16; ++i) {
      float4 vv = *(const float4*)&Vs[m][i * 4];
      acc[i].x += pp * vv.x; acc[i].y += pp * vv.y;
      acc[i].z += pp * vv.z; acc[i].w += pp * vv.w;
    }
  }
  const float inv = 1.0f / rowsum[(size_t)bh * NOUT + n];
  float* orow = &O[(size_t)(b * NOUT + n) * (SHEADS * SVD) + h * SVD];
#pragma unroll
  for (int i = 0; i < 16; ++i) {
    float4 o4 = make_float4(acc[i].x * inv, acc[i].y * inv,
                            acc[i].z * inv, acc[i].w * inv);
    *(float4*)&orow[i * 4] = o4;
  }
}

// ---------------------------------------------------------------------------
// Host side
// ---------------------------------------------------------------------------
struct BN {
  const float *W, *g, *b, *m, *v;
  int dout, din;
  float *Wf, *beta;
};

static BN getBN(void* const* d_in, int s, bool sorted, int dout, int din) {
  BN r;
  r.W = (const float*)d_in[s];
  if (sorted) {  // sorted keys: W,b,g,m,v
    r.b = (const float*)d_in[s + 1];
    r.g = (const float*)d_in[s + 2];
  } else {       // insertion: W,g,b,m,v
    r.g = (const float*)d_in[s + 1];
    r.b = (const float*)d_in[s + 2];
  }
  r.m = (const float*)d_in[s + 3];
  r.v = (const float*)d_in[s + 4];
  r.dout = dout; r.din = din; r.Wf = nullptr; r.beta = nullptr;
  return r;
}

static void fuse(BN& L, float*& wfp, float*& bp, hipStream_t st) {
  L.Wf = wfp;  wfp += (size_t)L.dout * L.din;
  L.beta = bp; bp  += L.dout;
  int tot = L.dout * L.din;
  fuse_bn<<<(tot + 255) / 256, 256, 0, st>>>(L.W, L.g, L.b, L.m, L.v, L.Wf,
                                             L.beta, L.dout, L.din);
}

template <bool HSW, bool RESID>
static void launch_gemm(const float* A, const BN& L, const float* R, float* D,
                        int M, hipStream_t st) {
  dim3 grid(M / 64, L.dout / 128);
  gemm_bn<HSW, RESID><<<grid, 256, 0, st>>>(A, L.Wf, L.beta, R, D, M, L.dout,
                                            L.din);
}

extern "C" void kernel_launch(void* const* d_in, const int* in_sizes, int n_in,
                              void* d_out, int out_size, void* d_ws,
                              size_t ws_size, hipStream_t stream) {
  (void)n_in; (void)out_size; (void)ws_size;
  // sorted-key pytree => leaf 1 is blocks[0].attn.bias (784);
  // insertion order  => blocks[0].attn.qkv.W (131072).
  const bool sorted = (in_sizes[1] == 784);

  BN qkv[4], proj[4], up[4], down[4];
  const float* abias[4];
  BN skv, sq, sproj, up2, down2;
  const float* sbias;
  for (int i = 0; i < 4; ++i) {
    int base = 1 + i * 21;
    if (sorted) {  // attn{bias,proj,qkv}, mlp{down,up}
      abias[i] = (const float*)d_in[base + 0];
      proj[i]  = getBN(d_in, base + 1, sorted, HID, HEADS * VDIM);
      qkv[i]   = getBN(d_in, base + 6, sorted, QKV_OUT, HID);
      down[i]  = getBN(d_in, base + 11, sorted, HID, HID * 2);
      up[i]    = getBN(d_in, base + 16, sorted, HID * 2, HID);
    } else {       // attn{qkv,proj,bias}, mlp{up,down}
      qkv[i]   = getBN(d_in, base + 0, sorted, QKV_OUT, HID);
      proj[i]  = getBN(d_in, base + 5, sorted, HID, HEADS * VDIM);
      abias[i] = (const float*)d_in[base + 10];
      up[i]    = getBN(d_in, base + 11, sorted, HID * 2, HID);
      down[i]  = getBN(d_in, base + 16, sorted, HID, HID * 2);
    }
  }
  if (sorted) {    // params{blocks, mlp2{down,up}, sub{bias,kv,proj,q}}
    down2 = getBN(d_in, 85, sorted, HID2, HID2 * 2);
    up2   = getBN(d_in, 90, sorted, HID2 * 2, HID2);
    sbias = (const float*)d_in[95];
    skv   = getBN(d_in, 96, sorted, KV_OUT, HID);
    sproj = getBN(d_in, 101, sorted, HID2, SHEADS * SVD);
    sq    = getBN(d_in, 106, sorted, SHEADS * SKD, HID);
  } else {         // params{blocks, sub{kv,q,proj,bias}, mlp2{up,down}}
    skv   = getBN(d_in, 85, sorted, KV_OUT, HID);
    sq    = getBN(d_in, 90, sorted, SHEADS * SKD, HID);
    sproj = getBN(d_in, 95, sorted, HID2, SHEADS * SVD);
    sbias = (const float*)d_in[100];
    up2   = getBN(d_in, 101, sorted, HID2 * 2, HID2);
    down2 = getBN(d_in, 106, sorted, HID2, HID2 * 2);
  }

  // --- workspace arenas (floats) ------------------------------------------
  float* ws = (float*)d_ws;
  size_t off = 0;
  auto arena = [&](size_t n) { float* p = ws + off; off += n; return p; };
  float* X    = arena((size_t)MT * HID);
  float* BIG  = arena((size_t)MT * KV_OUT);          // qkv / up / kv / up2
  float* S    = arena((size_t)BATCH * HEADS * NTOK * NTOK);
  float* O    = arena((size_t)MT * HID);
  float* RS   = arena((size_t)BATCH * SHEADS * NTOK);
  float* XQ   = arena((size_t)MT2 * HID);
  float* Q    = arena((size_t)MT2 * SHEADS * SKD);
  float* O2   = arena((size_t)MT2 * SHEADS * SVD);
  float* X2   = arena((size_t)MT2 * HID2);
  float* wfp  = arena(3000000);
  float* bp   = wfp + 2900000;

  for (int i = 0; i < 4; ++i) {
    fuse(qkv[i], wfp, bp, stream);
    fuse(proj[i], wfp, bp, stream);
    fuse(up[i], wfp, bp, stream);
    fuse(down[i], wfp, bp, stream);
  }
  fuse(skv, wfp, bp, stream);
  fuse(sq, wfp, bp, stream);
  fuse(sproj, wfp, bp, stream);
  fuse(up2, wfp, bp, stream);
  fuse(down2, wfp, bp, stream);

  hipMemcpyAsync(X, d_in[0], (size_t)MT * HID * sizeof(float),
                 hipMemcpyDeviceToDevice, stream);

  for (int i = 0; i < 4; ++i) {
    launch_gemm<false, false>(X, qkv[i], nullptr, BIG, MT, stream);
    attn_scores_main<<<BATCH * HEADS, 256, 0, stream>>>(BIG, abias[i], S, RS);
    attn_av_main<<<BATCH * HEADS, 256, 0, stream>>>(BIG, S, RS, O);
    launch_gemm<true, true>(O, proj[i], X, X, MT, stream);   // x += proj(hsw(o))
    launch_gemm<false, false>(X, up[i], nullptr, BIG, MT, stream);
    launch_gemm<true, true>(BIG, down[i], X, X, MT, stream); // x += down(hsw(up))
  }

  launch_gemm<false, false>(X, skv, nullptr, BIG, MT, stream);       // kv
  gather_q<<<(MT2 * (HID / 4) + 255) / 256, 256, 0, stream>>>(X, XQ);
  launch_gemm<false, false>(XQ, sq, nullptr, Q, MT2, stream);        // q
  attn_scores_sub<<<BATCH * SHEADS, 64, 0, stream>>>(Q, BIG, sbias, S, RS);
  attn_av_sub<<<BATCH * SHEADS, 64, 0, stream>>>(BIG, S, RS, O2);
  launch_gemm<true, false>(O2, sproj, nullptr, X2, MT2, stream);     // x2

  launch_gemm<false, false>(X2, up2, nullptr, BIG, MT2, stream);
  launch_gemm<true, true>(BIG, down2, X2, (float*)d_out, MT2, stream);
}